// WCE_JReg_71536975282965
// MI455X (gfx1250) — compile-verified
//
#include <hip/hip_runtime.h>
#include <hip/hip_bf16.h>
#include <math.h>

// ---------------------------------------------------------------------------
// WCE + Jaccard-style regularizer for MI455X (gfx1250), wave32.
//   input  : [8, 19, 512, 512] fp32     target : [8, 512, 512] int64
//   out    : scalar fp32
// Bandwidth-bound (~175 MB @ 23.3 TB/s ~= 7.5us floor). Single streaming
// pass; final_mat (19x19 = P^T Y) via V_WMMA_F32_16X16X4_F32 (fp32 MACs),
// with 1/S folded into the one-hot A operand. Raw v_exp/v_log/v_rcp
// (via builtins) strip libm denorm-fixup VALU so compute stays near the
// memory floor.
// ---------------------------------------------------------------------------

#define NCLS   19
#define HW     (512 * 512)
#define NPIX   (8 * HW)          // 2,097,152 pixels
#define GROUPS 8                 // 32-pixel groups per wave
#define NBLK   (NPIX / (32 * GROUPS))   // 8192 single-wave blocks

typedef float v2f __attribute__((ext_vector_type(2)));
typedef float v8f __attribute__((ext_vector_type(8)));

// d_ws float layout: [0..360] final_mat (row = channel i, col = class j),
//                    [361..379] per-class counts (float, exact: +1.0 steps),
//                    [380] loss accumulator
__global__ void wce_zero(float* __restrict__ ws) {
    int i = blockIdx.x * blockDim.x + threadIdx.x;
    if (i < NCLS * NCLS + NCLS + 1) ws[i] = 0.0f;
}

__global__ __launch_bounds__(32)
void wce_main(const float* __restrict__ x, const int* __restrict__ tgt,
              float* __restrict__ fm, float* __restrict__ cnt,
              float* __restrict__ lossAcc)
{
    __shared__ float lds_e[32 * 20];   // exp values, padded stride 20
    __shared__ float lds_r[32];        // 1/S per pixel
    __shared__ int   lds_t[32];        // class per pixel
    __shared__ float lds_c[NCLS];      // per-block class histogram

    const int lane = threadIdx.x;
    const int jl   = lane & 15;        // N/M index within a tile
    const int half = lane >> 4;        // 0: K=0,1   1: K=2,3

    if (lane < NCLS) lds_c[lane] = 0.0f;
    __syncthreads();

    v8f acc00 = {}, acc01 = {}, acc10 = {}, acc11 = {};
    float lsum = 0.0f;

    for (int g = 0; g < GROUPS; ++g) {
        const int pix = (blockIdx.x * GROUPS + g) * 32 + lane;
        const int n   = pix >> 18;            // / HW
        const int hw  = pix & (HW - 1);
        const float* px = x + (size_t)n * NCLS * HW + hw;

        int t = tgt[(size_t)2 * pix];         // low dword of little-endian i64
        t = t < 0 ? 0 : (t > NCLS - 1 ? NCLS - 1 : t);

        // ---- per-pixel softmax over 19 channels (coalesced strided loads) ----
        float xv[NCLS];
        float m = -3.4e38f;
        #pragma unroll
        for (int c = 0; c < NCLS; ++c) {
            xv[c] = px[(size_t)c * HW];
            m = fmaxf(m, xv[c]);
        }
        float S = 0.0f, xt = 0.0f;
        #pragma unroll
        for (int c = 0; c < NCLS; ++c) {
            // arg <= 0, raw v_exp_f32 safe (underflow -> 0, same as softmax)
            float e = __builtin_amdgcn_exp2f((xv[c] - m) * 1.44269504088896340f);
            S += e;
            lds_e[lane * 20 + c] = e;
            xt = (c == t) ? xv[c] : xt;
        }
        // S >= 1 (contains exp(0)) -> raw v_log_f32 (log2) is exact enough.
        // log_softmax(x+eps) == log_softmax(x): eps cancels exactly.
        lsum += (m + __builtin_amdgcn_logf(S) * 0.69314718055994531f) - xt;
        lds_r[lane] = __builtin_amdgcn_rcpf(S);
        lds_t[lane] = t;
        atomicAdd(&lds_c[t], 1.0f);
        __syncthreads();

        // ---- final_mat^T accumulation: D = (Y/S)^T * E, K in chunks of 4 ----
        #pragma unroll
        for (int kb = 0; kb < 8; ++kb) {
            const int p0 = kb * 4 + half * 2;   // this half-wave's K pair
            const int p1 = p0 + 1;
            const int t0 = lds_t[p0], t1 = lds_t[p1];
            const float r0 = lds_r[p0], r1 = lds_r[p1];

            v2f a0, a1, b0, b1;
            a0.x = (t0 == jl)       ? r0 : 0.0f;  a0.y = (t1 == jl)       ? r1 : 0.0f;
            a1.x = (t0 == jl + 16)  ? r0 : 0.0f;  a1.y = (t1 == jl + 16)  ? r1 : 0.0f;
            b0.x = lds_e[p0 * 20 + jl];           b0.y = lds_e[p1 * 20 + jl];
            const bool hi = (jl + 16) < NCLS;
            b1.x = hi ? lds_e[p0 * 20 + jl + 16] : 0.0f;
            b1.y = hi ? lds_e[p1 * 20 + jl + 16] : 0.0f;

            acc00 = __builtin_amdgcn_wmma_f32_16x16x4_f32(false, a0, false, b0,
                                                          (short)0, acc00, false, false);
            acc01 = __builtin_amdgcn_wmma_f32_16x16x4_f32(false, a0, false, b1,
                                                          (short)0, acc01, false, false);
            acc10 = __builtin_amdgcn_wmma_f32_16x16x4_f32(false, a1, false, b0,
                                                          (short)0, acc10, false, false);
            acc11 = __builtin_amdgcn_wmma_f32_16x16x4_f32(false, a1, false, b1,
                                                          (short)0, acc11, false, false);
        }
        __syncthreads();   // protect LDS reuse next group
    }

    // ---- loss: wave32 shuffle reduce, one atomic per wave ----
    #pragma unroll
    for (int off = 16; off > 0; off >>= 1) lsum += __shfl_xor(lsum, off, 32);
    if (lane == 0) atomicAdd(lossAcc, lsum);
    if (lane < NCLS) atomicAdd(&cnt[lane], lds_c[lane]);

    // ---- flush D tiles: D[m][n] = final_mat^T[class][chan] ----
    // 16x16 f32 C/D layout: VGPR r -> M = r + 8*half, N = lane&15
    #pragma unroll
    for (int r = 0; r < 8; ++r) {
        const int mrow = r + 8 * half;
        { int cls = mrow,      ch = jl;
          atomicAdd(&fm[ch * NCLS + cls], acc00[r]); }
        { int cls = mrow,      ch = jl + 16;
          if (ch < NCLS)                  atomicAdd(&fm[ch * NCLS + cls], acc01[r]); }
        { int cls = mrow + 16, ch = jl;
          if (cls < NCLS)                 atomicAdd(&fm[ch * NCLS + cls], acc10[r]); }
        { int cls = mrow + 16, ch = jl + 16;
          if (cls < NCLS && ch < NCLS)    atomicAdd(&fm[ch * NCLS + cls], acc11[r]); }
    }
}

__global__ __launch_bounds__(512)
void wce_final(const float* __restrict__ fm, const float* __restrict__ cnt,
               const float* __restrict__ lossAcc, float* __restrict__ out)
{
    __shared__ float red[512];
    const int tid = threadIdx.x;
    float term = 0.0f;
    if (tid < NCLS * NCLS) {
        const int i = tid / NCLS, j = tid % NCLS;    // i,j index classes
        const float ni = cnt[i], nj = cnt[j];
        const float sni = ni > 0.0f ? ni : 1.0f;
        const float snj = nj > 0.0f ? nj : 1.0f;
        const float ratio = 0.5f * (fm[i * NCLS + i] / sni)
                          - 0.5f * (fm[i * NCLS + j] / snj) + 0.5f;
        const bool valid = (ni * nj > 0.0f) && (i != j);
        const float sr = valid ? ratio : 1.0f;       // POWER == 1
        term = valid ? (-0.5f * logf(sr + 1e-8f)) : 0.0f;
    }
    red[tid] = term;
    __syncthreads();
    for (int s = 256; s > 0; s >>= 1) {
        if (tid < s) red[tid] += red[tid + s];
        __syncthreads();
    }
    if (tid == 0)
        out[0] = lossAcc[0] * (1.0f / (float)NPIX) + red[0];
}

extern "C" void kernel_launch(void* const* d_in, const int* in_sizes, int n_in,
                              void* d_out, int out_size, void* d_ws, size_t ws_size,
                              hipStream_t stream) {
    const float* x   = (const float*)d_in[0];
    const int*   tgt = (const int*)d_in[1];     // int64 targets, read low dwords
    float* out = (float*)d_out;

    float* fm      = (float*)d_ws;              // 361 floats
    float* cnt     = fm + NCLS * NCLS;          // 19 floats
    float* lossAcc = cnt + NCLS;                // 1 float

    wce_zero <<<1, 512, 0, stream>>>(fm);
    wce_main <<<NBLK, 32, 0, stream>>>(x, tgt, fm, cnt, lossAcc);
    wce_final<<<1, 512, 0, stream>>>(fm, cnt, lossAcc, out);
}